// KabschRMSD_66881230733429
// MI455X (gfx1250) — compile-verified
//
#include <hip/hip_runtime.h>
#include <math.h>

typedef float v2f __attribute__((ext_vector_type(2)));
typedef float v8f __attribute__((ext_vector_type(8)));

#define NPTS 512
#define TPB  256
#define NWAVES 8                      // 256 threads / wave32
#define PTS_PER_WAVE (NPTS / NWAVES)  // 64
#define CHUNKS (PTS_PER_WAVE / 4)     // 16 WMMA (K=4) steps per wave

// One block per batch element. Coordinates staged in LDS once; covariance via
// v_wmma_f32_16x16x4_f32 (M,N padded 3->16, extra ones-row/col folds the
// coordinate sums into the same accumulator), SVD on 3x3 by thread 0,
// RMSD pass from LDS, deterministic tree reductions throughout.
__global__ __launch_bounds__(TPB) void kabsch_rmsd_kernel(
    const float* __restrict__ pred,   // [B, 512, 3]
    const float* __restrict__ refc,   // [B, 512, 3]
    float* __restrict__ rmsd_out)     // [B]
{
    __shared__ float sp[NPTS * 3];        // pred coords
    __shared__ float sq[NPTS * 3];        // ref coords
    __shared__ float wpart[NWAVES][4][4]; // per-wave WMMA partials (top-left 4x4)
    __shared__ float red[TPB];            // reduction scratch
    __shared__ float sRt[12];             // R row-major (9) + t (3)

    const int b   = blockIdx.x;
    const int tid = threadIdx.x;
    const float* gp = pred + (size_t)b * (NPTS * 3);
    const float* gq = refc + (size_t)b * (NPTS * 3);

    // ---- stage both point clouds into LDS (1536 floats = 384 float4 each) ----
    {
        const float4* gp4 = (const float4*)gp;
        const float4* gq4 = (const float4*)gq;
        float4* sp4 = (float4*)sp;
        float4* sq4 = (float4*)sq;
        sp4[tid] = gp4[tid];
        sq4[tid] = gq4[tid];
        if (tid < 128) {
            sp4[tid + TPB] = gp4[tid + TPB];
            sq4[tid + TPB] = gq4[tid + TPB];
        }
    }
    __syncthreads();

    // ---- WMMA: D[i][j] = sum_n Ahat[n][i] * Bhat[n][j] over this wave's K range
    // Ahat[n][i] = P[n][i] for i<3, 1.0 for i==3  (col 3 of D = sum P)
    // Bhat[n][j] = Q[n][j] for j<3, 1.0 for j==3  (row 3 of D = sum Q)
    // D[0:3][0:3] = raw cross-product sums, D[3][3] = point count.
    //
    // Branch-free fragments: every lane does an unconditional LDS load from a
    // safe (broadcast) address, then frag = load*msk + pad. EXEC stays all-1s.
    const int wave = tid >> 5;
    const int lane = tid & 31;
    const int m    = lane & 15;
    const int hi   = lane >> 4;     // 0: K0/K1 half, 1: K2/K3 half (32-bit layout)
    {
        const int   mc  = (m < 3) ? m : 0;     // safe coord index (broadcast)
        const float msk = (m < 3) ? 1.0f : 0.0f;
        const float pad = (m == 3) ? 1.0f : 0.0f;

        v8f acc = {0.f, 0.f, 0.f, 0.f, 0.f, 0.f, 0.f, 0.f};
        // float index of the K=2*hi point of this wave's chunk 0, coord mc
        int off = (wave * PTS_PER_WAVE + 2 * hi) * 3 + mc;
#pragma unroll
        for (int c = 0; c < CHUNKS; ++c) {
            float a0 = sp[off], a1 = sp[off + 3];   // points n0, n0+1
            float b0 = sq[off], b1 = sq[off + 3];
            v2f av, bv;
            av.x = fmaf(a0, msk, pad);   // A: K = 2*hi
            av.y = fmaf(a1, msk, pad);   // A: K = 2*hi + 1
            bv.x = fmaf(b0, msk, pad);   // B: K = 2*hi
            bv.y = fmaf(b1, msk, pad);   // B: K = 2*hi + 1
            acc = __builtin_amdgcn_wmma_f32_16x16x4_f32(
                false, av, false, bv, (short)0, acc, false, false);
            off += 12;                   // advance 4 points (4 * 3 floats)
        }
        // C/D layout: VGPR i holds row M=i for lanes 0-15 (N = lane).
        if (lane < 4) {
            wpart[wave][0][lane] = acc[0];
            wpart[wave][1][lane] = acc[1];
            wpart[wave][2][lane] = acc[2];
            wpart[wave][3][lane] = acc[3];
        }
    }
    __syncthreads();

    // ---- thread 0: assemble covariance, 3x3 SVD (Jacobi on A^T A), build R,t ----
    if (tid == 0) {
        float T[4][4];
        for (int i = 0; i < 4; ++i)
            for (int j = 0; j < 4; ++j) {
                float s = 0.f;
                for (int w = 0; w < NWAVES; ++w) s += wpart[w][i][j];  // fixed order
                T[i][j] = s;
            }
        const float invN = 1.0f / (float)NPTS;
        float mp[3], mq[3];
        for (int i = 0; i < 3; ++i) { mp[i] = T[i][3] * invN; mq[i] = T[3][i] * invN; }

        float A[3][3];
        for (int i = 0; i < 3; ++i)
            for (int j = 0; j < 3; ++j)
                A[i][j] = T[i][j] - T[i][3] * T[3][j] * invN;   // Spq - Sp Sq^T / N

        float det = A[0][0] * (A[1][1] * A[2][2] - A[1][2] * A[2][1])
                  - A[0][1] * (A[1][0] * A[2][2] - A[1][2] * A[2][0])
                  + A[0][2] * (A[1][0] * A[2][1] - A[1][1] * A[2][0]);
        const float dsgn = (det < 0.f) ? -1.f : 1.f;

        // M = A^T A (symmetric PSD)
        float M[3][3];
        for (int i = 0; i < 3; ++i)
            for (int j = 0; j < 3; ++j)
                M[i][j] = A[0][i] * A[0][j] + A[1][i] * A[1][j] + A[2][i] * A[2][j];

        // cyclic Jacobi eigendecomposition: M = V diag(w) V^T
        float V[3][3] = {{1.f, 0.f, 0.f}, {0.f, 1.f, 0.f}, {0.f, 0.f, 1.f}};
        for (int sweep = 0; sweep < 12; ++sweep) {
            float offd = fabsf(M[0][1]) + fabsf(M[0][2]) + fabsf(M[1][2]);
            float dia  = fabsf(M[0][0]) + fabsf(M[1][1]) + fabsf(M[2][2]);
            if (offd <= 1e-10f * (dia + 1e-30f)) break;
            for (int pi = 0; pi < 3; ++pi) {
                const int p = (pi == 2) ? 1 : 0;
                const int q = (pi == 0) ? 1 : 2;
                const int r = 3 - p - q;
                float apq = M[p][q];
                if (fabsf(apq) < 1e-30f) continue;
                float theta = (M[q][q] - M[p][p]) / (2.f * apq);
                float tt = copysignf(1.f, theta) /
                           (fabsf(theta) + sqrtf(theta * theta + 1.f));
                float cc = 1.f / sqrtf(tt * tt + 1.f);
                float ss = tt * cc;
                M[p][p] -= tt * apq;
                M[q][q] += tt * apq;
                M[p][q] = M[q][p] = 0.f;
                float mrp = M[r][p], mrq = M[r][q];
                M[r][p] = M[p][r] = cc * mrp - ss * mrq;
                M[r][q] = M[q][r] = ss * mrp + cc * mrq;
                for (int i = 0; i < 3; ++i) {
                    float vp = V[i][p], vq = V[i][q];
                    V[i][p] = cc * vp - ss * vq;
                    V[i][q] = ss * vp + cc * vq;
                }
            }
        }
        float wv[3] = {M[0][0], M[1][1], M[2][2]};
        // sort eigenpairs descending (SVD convention; flip goes on smallest)
        for (int i = 0; i < 2; ++i)
            for (int j = i + 1; j < 3; ++j)
                if (wv[j] > wv[i]) {
                    float t0 = wv[i]; wv[i] = wv[j]; wv[j] = t0;
                    for (int k = 0; k < 3; ++k) {
                        float tv = V[k][i]; V[k][i] = V[k][j]; V[k][j] = tv;
                    }
                }
        // U columns: u_k = A v_k / sigma_k
        float U[3][3];
        for (int k = 0; k < 3; ++k) {
            float sk = sqrtf(fmaxf(wv[k], 0.f));
            float inv = 1.f / fmaxf(sk, 1e-12f);
            for (int i = 0; i < 3; ++i)
                U[i][k] = (A[i][0] * V[0][k] + A[i][1] * V[1][k] + A[i][2] * V[2][k]) * inv;
        }
        // R = U diag(1,1,d) V^T ; t = mp - R mq
        float R[3][3];
        for (int i = 0; i < 3; ++i)
            for (int j = 0; j < 3; ++j)
                R[i][j] = U[i][0] * V[j][0] + U[i][1] * V[j][1] + dsgn * U[i][2] * V[j][2];
        for (int i = 0; i < 3; ++i) {
            for (int j = 0; j < 3; ++j) sRt[i * 3 + j] = R[i][j];
            sRt[9 + i] = mp[i] - (R[i][0] * mq[0] + R[i][1] * mq[1] + R[i][2] * mq[2]);
        }
    }
    __syncthreads();

    // ---- RMSD pass from LDS ----
    {
        const float R00 = sRt[0], R01 = sRt[1], R02 = sRt[2];
        const float R10 = sRt[3], R11 = sRt[4], R12 = sRt[5];
        const float R20 = sRt[6], R21 = sRt[7], R22 = sRt[8];
        const float t0 = sRt[9], t1 = sRt[10], t2 = sRt[11];
        float local = 0.f;
        for (int n = tid; n < NPTS; n += TPB) {
            float qx = sq[n * 3 + 0], qy = sq[n * 3 + 1], qz = sq[n * 3 + 2];
            float px = sp[n * 3 + 0], py = sp[n * 3 + 1], pz = sp[n * 3 + 2];
            float ax = R00 * qx + R01 * qy + R02 * qz + t0;
            float ay = R10 * qx + R11 * qy + R12 * qz + t1;
            float az = R20 * qx + R21 * qy + R22 * qz + t2;
            float dx = px - ax, dy = py - ay, dz = pz - az;
            local += dx * dx + dy * dy + dz * dz;
        }
        red[tid] = local;
    }
    __syncthreads();
    for (int s = TPB / 2; s > 0; s >>= 1) {
        if (tid < s) red[tid] += red[tid + s];
        __syncthreads();
    }
    if (tid == 0) rmsd_out[b] = sqrtf(red[0] / (float)NPTS);
}

// Deterministic mean over B per-batch RMSDs.
__global__ __launch_bounds__(256) void rmsd_mean_kernel(
    const float* __restrict__ ws, float* __restrict__ out, int B)
{
    __shared__ float red[256];
    const int tid = threadIdx.x;
    float s = 0.f;
    for (int i = tid; i < B; i += 256) s += ws[i];   // fixed-order strided sum
    red[tid] = s;
    __syncthreads();
    for (int st = 128; st > 0; st >>= 1) {
        if (tid < st) red[tid] += red[tid + st];
        __syncthreads();
    }
    if (tid == 0) out[0] = red[0] / (float)B;
}

extern "C" void kernel_launch(void* const* d_in, const int* in_sizes, int n_in,
                              void* d_out, int out_size, void* d_ws, size_t ws_size,
                              hipStream_t stream) {
    const float* pred = (const float*)d_in[0];   // ligs_coords_pred [B,512,3]
    const float* refc = (const float*)d_in[1];   // ligs_coords      [B,512,3]
    float* out = (float*)d_out;                  // scalar f32
    float* ws  = (float*)d_ws;                   // B floats of scratch

    const int B = in_sizes[0] / (NPTS * 3);      // 8192 for the reference shapes

    kabsch_rmsd_kernel<<<B, TPB, 0, stream>>>(pred, refc, ws);
    rmsd_mean_kernel<<<1, 256, 0, stream>>>(ws, out, B);
}